// QuadProposalModule_61306363183176
// MI455X (gfx1250) — compile-verified
//
#include <hip/hip_runtime.h>
#include <math.h>

#define H_DIM   256
#define P_DIM   1024
#define SMAX    4000
#define KNRM    20
#define KSEL    10
#define RAD2    0.04f
#define BN_EPS  1e-5f

typedef float v2f __attribute__((ext_vector_type(2)));
typedef float v8f __attribute__((ext_vector_type(8)));

// ------------------------------------------------------------------
// GEMM via V_WMMA_F32_16X16X4_F32:  Y[b,m,n] = sum_k W[m,k]*X[b,k,n] + bias[m]
// One wave32 computes a 16x64 strip (4 tiles) sharing one A fragment:
// 4 independent accumulators break the WMMA C-dependency chain and
// quarter the A-side load traffic.
// ------------------------------------------------------------------
__global__ __launch_bounds__(256) void k_gemm_wmma(
    const float* __restrict__ X, const float* __restrict__ W,
    const float* __restrict__ bias, float* __restrict__ Y,
    int B, int M, int N, int K)
{
  const int lane = threadIdx.x & 31;
  const int wid  = (blockIdx.x * blockDim.x + threadIdx.x) >> 5;
  const int tilesN4 = N >> 6;           // groups of 4 adjacent 16-wide tiles
  const int tilesM  = M >> 4;
  const int tpb = tilesM * tilesN4;
  const int b = wid / tpb;
  if (b >= B) return;                   // wave-uniform branch (EXEC stays all-1s)
  const int t  = wid - b * tpb;
  const int tm = t / tilesN4, tg = t - tm * tilesN4;
  const int lrow  = lane & 15;          // A: M row within tile; B: N col within tile
  const int khalf = (lane >> 4) << 1;   // 0 or 2

  const float* Wrow  = W + (size_t)(tm * 16 + lrow) * K;
  const float* Xbase = X + (size_t)b * K * N + (size_t)(tg * 64 + lrow);

  v8f acc0 = {}, acc1 = {}, acc2 = {}, acc3 = {};
  for (int k0 = 0; k0 < K; k0 += 4) {
    v2f a;
    a.x = Wrow[k0 + khalf + 0];
    a.y = Wrow[k0 + khalf + 1];
    const float* x0 = Xbase + (size_t)(k0 + khalf + 0) * N;
    const float* x1 = Xbase + (size_t)(k0 + khalf + 1) * N;
    v2f b0, b1, b2, b3;
    b0.x = x0[0];  b0.y = x1[0];
    b1.x = x0[16]; b1.y = x1[16];
    b2.x = x0[32]; b2.y = x1[32];
    b3.x = x0[48]; b3.y = x1[48];
    acc0 = __builtin_amdgcn_wmma_f32_16x16x4_f32(false, a, false, b0, (short)0, acc0, false, false);
    acc1 = __builtin_amdgcn_wmma_f32_16x16x4_f32(false, a, false, b1, (short)0, acc1, false, false);
    acc2 = __builtin_amdgcn_wmma_f32_16x16x4_f32(false, a, false, b2, (short)0, acc2, false, false);
    acc3 = __builtin_amdgcn_wmma_f32_16x16x4_f32(false, a, false, b3, (short)0, acc3, false, false);
  }
  const int rbase = tm * 16 + ((lane >> 4) << 3);
  float* Yout = Y + (size_t)b * M * N + (size_t)(tg * 64 + lrow);
#pragma unroll
  for (int r = 0; r < 8; ++r) {
    const int row = rbase + r;
    const float bv = bias[row];
    float* yr = Yout + (size_t)row * N;
    yr[0]  = acc0[r] + bv;
    yr[16] = acc1[r] + bv;
    yr[32] = acc2[r] + bv;
    yr[48] = acc3[r] + bv;
  }
}

// ------------------------------------------------------------------
// BatchNorm training-mode stats per channel over (B, N)
// ------------------------------------------------------------------
__global__ __launch_bounds__(256) void k_bnstats(
    const float* __restrict__ Y, float* __restrict__ stats, int B, int C, int N)
{
  const int c = blockIdx.x;
  __shared__ float sS[256], sQ[256];
  float s = 0.f, q = 0.f;
  const int total = B * N;
  for (int i = threadIdx.x; i < total; i += 256) {
    int b = i / N, p = i - b * N;
    float v = Y[(size_t)b * C * N + (size_t)c * N + p];
    s += v; q += v * v;
  }
  sS[threadIdx.x] = s; sQ[threadIdx.x] = q;
  __syncthreads();
  for (int off = 128; off > 0; off >>= 1) {
    if (threadIdx.x < off) {
      sS[threadIdx.x] += sS[threadIdx.x + off];
      sQ[threadIdx.x] += sQ[threadIdx.x + off];
    }
    __syncthreads();
  }
  if (threadIdx.x == 0) {
    float inv_n = 1.0f / (float)total;
    float mu  = sS[0] * inv_n;
    float var = sQ[0] * inv_n - mu * mu;      // biased variance (matches ref)
    stats[2 * c + 0] = mu;
    stats[2 * c + 1] = rsqrtf(var + BN_EPS);
  }
}

__global__ void k_bnrelu(
    const float* __restrict__ Y, const float* __restrict__ stats,
    const float* __restrict__ g, const float* __restrict__ be,
    float* __restrict__ O, int B, int C, int N)
{
  size_t i = (size_t)blockIdx.x * blockDim.x + threadIdx.x;
  size_t tot = (size_t)B * C * N;
  if (i >= tot) return;
  int c = (int)((i / (size_t)N) % (size_t)C);
  float y = (Y[i] - stats[2 * c]) * stats[2 * c + 1];
  y = y * g[c] + be[c];
  O[i] = fmaxf(y, 0.f);
}

// ------------------------------------------------------------------
// Heads: 7 output channels (qs2 | center3(+base_xyz) | size2)
// ------------------------------------------------------------------
__global__ __launch_bounds__(256) void k_heads(
    const float* __restrict__ X,
    const float* __restrict__ Wqs, const float* __restrict__ bqs,
    const float* __restrict__ Wc,  const float* __restrict__ bc,
    const float* __restrict__ Wsz, const float* __restrict__ bsz,
    const float* __restrict__ base_xyz, float* __restrict__ out,
    float* __restrict__ ctrs, int B, int N)
{
  __shared__ float sW[7 * H_DIM];
  for (int i = threadIdx.x; i < 7 * H_DIM; i += 256) {
    int j = i / H_DIM, c = i - j * H_DIM;
    float v = (j < 2) ? Wqs[j * H_DIM + c]
            : (j < 5) ? Wc[(j - 2) * H_DIM + c]
                      : Wsz[(j - 5) * H_DIM + c];
    sW[i] = v;
  }
  __syncthreads();
  int gid = blockIdx.x * 256 + threadIdx.x;
  if (gid >= B * N) return;
  int b = gid / N, p = gid - b * N;
  float acc[7] = {0.f, 0.f, 0.f, 0.f, 0.f, 0.f, 0.f};
  const float* Xp = X + (size_t)b * H_DIM * N + p;
  for (int c = 0; c < H_DIM; ++c) {
    float x = Xp[(size_t)c * N];
#pragma unroll
    for (int j = 0; j < 7; ++j) acc[j] += x * sW[j * H_DIM + c];
  }
  float* o = out + (size_t)gid * 10;
  o[0] = acc[0] + bqs[0];
  o[1] = acc[1] + bqs[1];
#pragma unroll
  for (int r = 0; r < 3; ++r) {
    float cr = acc[2 + r] + bc[r] + base_xyz[(size_t)gid * 3 + r];
    o[2 + r] = cr;
    ctrs[(size_t)gid * 3 + r] = cr;
  }
  o[5] = acc[5] + bsz[0];
  o[6] = acc[6] + bsz[1];
}

// ------------------------------------------------------------------
// Gather sampled points
// ------------------------------------------------------------------
__global__ void k_gather(const float* __restrict__ pc, const int* __restrict__ sidx,
                         float* __restrict__ pts, int B, int Npc, int S)
{
  int gid = blockIdx.x * blockDim.x + threadIdx.x;
  if (gid >= B * S) return;
  int b = gid / S;
  int idx = sidx[gid];
  const float* src = pc + ((size_t)b * Npc + (size_t)idx) * 3;
  float* dst = pts + (size_t)gid * 3;
  dst[0] = src[0]; dst[1] = src[1]; dst[2] = src[2];
}

// ------------------------------------------------------------------
// Smallest-eigenvalue eigenvector of symmetric 3x3 (sign arbitrary;
// resolved by the outward-orientation step afterwards)
// ------------------------------------------------------------------
__device__ inline void smallest_evec(float a00, float a01, float a02,
                                     float a11, float a12, float a22,
                                     float& nx, float& ny, float& nz)
{
  float p1 = a01 * a01 + a02 * a02 + a12 * a12;
  float q  = (a00 + a11 + a22) * (1.0f / 3.0f);
  float b00 = a00 - q, b11 = a11 - q, b22 = a22 - q;
  float p2 = b00 * b00 + b11 * b11 + b22 * b22 + 2.f * p1;
  if (p2 <= 1e-24f) { nx = 0.f; ny = 0.f; nz = 1.f; return; }
  float p = sqrtf(p2 * (1.0f / 6.0f));
  float inv = 1.0f / p;
  float c00 = b00 * inv, c01 = a01 * inv, c02 = a02 * inv;
  float c11 = b11 * inv, c12 = a12 * inv, c22 = b22 * inv;
  float detB = c00 * (c11 * c22 - c12 * c12)
             - c01 * (c01 * c22 - c12 * c02)
             + c02 * (c01 * c12 - c11 * c02);
  float r = fminf(1.f, fmaxf(-1.f, 0.5f * detB));
  float phi = acosf(r) * (1.0f / 3.0f);
  float lmin = q + 2.f * p * cosf(phi + 2.09439510239319549f);
  float m00 = a00 - lmin, m11 = a11 - lmin, m22 = a22 - lmin;
  float c0x = a01 * a12 - a02 * m11, c0y = a02 * a01 - m00 * a12, c0z = m00 * m11 - a01 * a01;
  float c1x = a01 * m22 - a02 * a12, c1y = a02 * a02 - m00 * m22, c1z = m00 * a12 - a01 * a02;
  float c2x = m11 * m22 - a12 * a12, c2y = a12 * a02 - a01 * m22, c2z = a01 * a12 - m11 * a02;
  float n0 = c0x * c0x + c0y * c0y + c0z * c0z;
  float n1 = c1x * c1x + c1y * c1y + c1z * c1z;
  float n2 = c2x * c2x + c2y * c2y + c2z * c2z;
  float bx = c0x, by = c0y, bz = c0z, bn = n0;
  if (n1 > bn) { bx = c1x; by = c1y; bz = c1z; bn = n1; }
  if (n2 > bn) { bx = c2x; by = c2y; bz = c2z; bn = n2; }
  if (bn < 1e-30f) { nx = 0.f; ny = 0.f; nz = 1.f; return; }
  float s = rsqrtf(bn);
  nx = bx * s; ny = by * s; nz = bz * s;
}

// ------------------------------------------------------------------
// Per-point PCA normals: batch point set lives in LDS (48KB of 320KB/WGP);
// top-20 NN kept sorted in registers (stable on ties -> lower index first).
// ------------------------------------------------------------------
__global__ __launch_bounds__(256) void k_normals(
    const float* __restrict__ pts, float* __restrict__ nrm, int S)
{
  __shared__ float sp[SMAX * 3];
  __shared__ float sred[256 * 3];
  const int b = blockIdx.y;
  const float* bp = pts + (size_t)b * S * 3;
  for (int i = threadIdx.x; i < S * 3; i += 256) sp[i] = bp[i];
  __syncthreads();

  // batch centroid (redundant per block, trivial cost, avoids global atomics)
  float cx = 0.f, cy = 0.f, cz = 0.f;
  for (int j = threadIdx.x; j < S; j += 256) {
    cx += sp[j * 3]; cy += sp[j * 3 + 1]; cz += sp[j * 3 + 2];
  }
  sred[threadIdx.x * 3 + 0] = cx;
  sred[threadIdx.x * 3 + 1] = cy;
  sred[threadIdx.x * 3 + 2] = cz;
  __syncthreads();
  for (int off = 128; off > 0; off >>= 1) {
    if (threadIdx.x < off) {
      sred[threadIdx.x * 3 + 0] += sred[(threadIdx.x + off) * 3 + 0];
      sred[threadIdx.x * 3 + 1] += sred[(threadIdx.x + off) * 3 + 1];
      sred[threadIdx.x * 3 + 2] += sred[(threadIdx.x + off) * 3 + 2];
    }
    __syncthreads();
  }
  const float invS = 1.0f / (float)S;
  const float ctrx = sred[0] * invS, ctry = sred[1] * invS, ctrz = sred[2] * invS;

  const int s = blockIdx.x * 256 + threadIdx.x;
  if (s >= S) return;
  const float px = sp[s * 3], py = sp[s * 3 + 1], pz = sp[s * 3 + 2];

  float bd[KNRM]; int bi[KNRM];
#pragma unroll
  for (int t = 0; t < KNRM; ++t) { bd[t] = 3.4e38f; bi[t] = 0; }
  for (int j = 0; j < S; ++j) {           // same j for all lanes -> LDS broadcast
    float dx = sp[j * 3] - px, dy = sp[j * 3 + 1] - py, dz = sp[j * 3 + 2] - pz;
    float d2 = dx * dx + dy * dy + dz * dz;
    if (d2 < bd[KNRM - 1]) {
      float cd = d2; int ci = j;
#pragma unroll
      for (int t = 0; t < KNRM; ++t) {
        if (cd < bd[t]) {
          float td = bd[t]; bd[t] = cd; cd = td;
          int ti = bi[t]; bi[t] = ci; ci = ti;
        }
      }
    }
  }
  // radius-masked weighted mean + covariance (w in {0,1}, w^2 == w)
  float sw = 0.f, mx = 0.f, my = 0.f, mz = 0.f;
#pragma unroll
  for (int t = 0; t < KNRM; ++t) {
    float w = (bd[t] <= RAD2) ? 1.f : 0.f;
    int j = bi[t];
    sw += w;
    mx += w * sp[j * 3]; my += w * sp[j * 3 + 1]; mz += w * sp[j * 3 + 2];
  }
  float icnt = 1.0f / fmaxf(sw, 1.f);
  mx *= icnt; my *= icnt; mz *= icnt;
  float a00 = 0.f, a01 = 0.f, a02 = 0.f, a11 = 0.f, a12 = 0.f, a22 = 0.f;
#pragma unroll
  for (int t = 0; t < KNRM; ++t) {
    float w = (bd[t] <= RAD2) ? 1.f : 0.f;
    int j = bi[t];
    float dx = (sp[j * 3] - mx) * w;
    float dy = (sp[j * 3 + 1] - my) * w;
    float dz = (sp[j * 3 + 2] - mz) * w;
    a00 += dx * dx; a01 += dx * dy; a02 += dx * dz;
    a11 += dy * dy; a12 += dy * dz; a22 += dz * dz;
  }
  float nx, ny, nz;
  smallest_evec(a00, a01, a02, a11, a12, a22, nx, ny, nz);
  // ref: flip if dot<0, then negate all => final = (dot<0) ? n : -n
  float dot = (px - ctrx) * nx + (py - ctry) * ny + (pz - ctrz) * nz;
  float sgn = (dot < 0.f) ? 1.f : -1.f;
  float* o = nrm + ((size_t)b * S + s) * 3;
  o[0] = nx * sgn; o[1] = ny * sgn; o[2] = nz * sgn;
}

// ------------------------------------------------------------------
// Per-proposal: top-10 nearest sampled points to center, average normals,
// z <- 0, normalize, nan -> 1e-6
// ------------------------------------------------------------------
__global__ __launch_bounds__(256) void k_select(
    const float* __restrict__ pts, const float* __restrict__ nrm,
    const float* __restrict__ ctrs, float* __restrict__ out, int S, int N)
{
  __shared__ float sp[SMAX * 3];
  const int b = blockIdx.y;
  const float* bp = pts + (size_t)b * S * 3;
  for (int i = threadIdx.x; i < S * 3; i += 256) sp[i] = bp[i];
  __syncthreads();

  const int p = blockIdx.x * 256 + threadIdx.x;
  if (p >= N) return;
  const size_t gp = (size_t)b * N + p;
  const float cx = ctrs[gp * 3], cy = ctrs[gp * 3 + 1], cz = ctrs[gp * 3 + 2];

  float bd[KSEL]; int bi[KSEL];
#pragma unroll
  for (int t = 0; t < KSEL; ++t) { bd[t] = 3.4e38f; bi[t] = 0; }
  for (int j = 0; j < S; ++j) {
    float dx = cx - sp[j * 3], dy = cy - sp[j * 3 + 1], dz = cz - sp[j * 3 + 2];
    float d2 = dx * dx + dy * dy + dz * dz;
    if (d2 < bd[KSEL - 1]) {
      float cd = d2; int ci = j;
#pragma unroll
      for (int t = 0; t < KSEL; ++t) {
        if (cd < bd[t]) {
          float td = bd[t]; bd[t] = cd; cd = td;
          int ti = bi[t]; bi[t] = ci; ci = ti;
        }
      }
    }
  }
  float nx = 0.f, ny = 0.f;
  const float* bn = nrm + (size_t)b * S * 3;
#pragma unroll
  for (int t = 0; t < KSEL; ++t) {
    int j = bi[t];
    nx += bn[j * 3]; ny += bn[j * 3 + 1];
  }
  nx *= (1.0f / KSEL); ny *= (1.0f / KSEL);         // z component zeroed by ref
  float nn = sqrtf(nx * nx + ny * ny);
  float sx = nx / nn, sy = ny / nn, sz = 0.0f / nn; // 0/0 -> NaN like the ref
  if (isnan(sx)) sx = 1e-6f;
  if (isnan(sy)) sy = 1e-6f;
  if (isnan(sz)) sz = 1e-6f;
  float* o = out + gp * 10;
  o[7] = sx; o[8] = sy; o[9] = sz;
}

// ------------------------------------------------------------------
extern "C" void kernel_launch(void* const* d_in, const int* in_sizes, int n_in,
                              void* d_out, int out_size, void* d_ws, size_t ws_size,
                              hipStream_t stream)
{
  (void)n_in; (void)out_size; (void)ws_size;
  const float* net      = (const float*)d_in[0];
  const float* base_xyz = (const float*)d_in[1];
  const float* pc       = (const float*)d_in[2];
  const int*   sidx     = (const int*)d_in[3];
  const float* W1  = (const float*)d_in[4];
  const float* b1  = (const float*)d_in[5];
  const float* g1  = (const float*)d_in[6];
  const float* be1 = (const float*)d_in[7];
  const float* W2  = (const float*)d_in[8];
  const float* b2  = (const float*)d_in[9];
  const float* g2  = (const float*)d_in[10];
  const float* be2 = (const float*)d_in[11];
  const float* Wqs = (const float*)d_in[12];
  const float* bqs = (const float*)d_in[13];
  const float* Wc  = (const float*)d_in[14];
  const float* bc  = (const float*)d_in[15];
  const float* Wsz = (const float*)d_in[16];
  const float* bsz = (const float*)d_in[17];
  float* out = (float*)d_out;

  const int B   = in_sizes[0] / (H_DIM * P_DIM);
  const int Npc = in_sizes[2] / (B * 3);
  const int S   = in_sizes[3] / B;

  // workspace layout (floats)
  float* ws    = (float*)d_ws;
  float* Ybuf  = ws;                                   // B*H*P
  float* Xbuf  = Ybuf + (size_t)B * H_DIM * P_DIM;     // B*H*P
  float* stats = Xbuf + (size_t)B * H_DIM * P_DIM;     // 2*H
  float* pts   = stats + 2 * H_DIM;                    // B*S*3
  float* nrms  = pts + (size_t)B * S * 3;              // B*S*3
  float* ctrs  = nrms + (size_t)B * S * 3;             // B*P*3

  const int waves = B * (H_DIM / 16) * (P_DIM / 64);   // 16x64 strip per wave
  const int gemmBlocks = (waves * 32 + 255) / 256;
  const size_t tot = (size_t)B * H_DIM * P_DIM;
  const int ewBlocks = (int)((tot + 255) / 256);

  // layer 1
  k_gemm_wmma<<<gemmBlocks, 256, 0, stream>>>(net, W1, b1, Ybuf, B, H_DIM, P_DIM, H_DIM);
  k_bnstats<<<H_DIM, 256, 0, stream>>>(Ybuf, stats, B, H_DIM, P_DIM);
  k_bnrelu<<<ewBlocks, 256, 0, stream>>>(Ybuf, stats, g1, be1, Xbuf, B, H_DIM, P_DIM);
  // layer 2
  k_gemm_wmma<<<gemmBlocks, 256, 0, stream>>>(Xbuf, W2, b2, Ybuf, B, H_DIM, P_DIM, H_DIM);
  k_bnstats<<<H_DIM, 256, 0, stream>>>(Ybuf, stats, B, H_DIM, P_DIM);
  k_bnrelu<<<ewBlocks, 256, 0, stream>>>(Ybuf, stats, g2, be2, Xbuf, B, H_DIM, P_DIM);
  // heads -> out[...,0:7] and centers
  k_heads<<<(B * P_DIM + 255) / 256, 256, 0, stream>>>(
      Xbuf, Wqs, bqs, Wc, bc, Wsz, bsz, base_xyz, out, ctrs, B, P_DIM);
  // geometry path
  k_gather<<<(B * S + 255) / 256, 256, 0, stream>>>(pc, sidx, pts, B, Npc, S);
  k_normals<<<dim3((S + 255) / 256, B), 256, 0, stream>>>(pts, nrms, S);
  k_select<<<dim3((P_DIM + 255) / 256, B), 256, 0, stream>>>(pts, nrms, ctrs, out, S, P_DIM);
}